// ChamferDistanceLoss_695784702577
// MI455X (gfx1250) — compile-verified
//
#include <hip/hip_runtime.h>

typedef __attribute__((ext_vector_type(2))) float v2f;
typedef __attribute__((ext_vector_type(8))) float v8f;

#define BATCH 16
#define NPT   4096

__global__ void chamfer_zero_kernel(float* out) { out[0] = 0.0f; }

// One wave computes dist-mins for a 16-row tile of X against all N points of Y.
//
// WMMA f32 16x16x4 with the padded K=4 slot used to fuse the row norm:
//   A[m,:] = [-2x0, -2x1, -2x2, ||x_m||^2]   (built once per row tile)
//   B[:,n] = [ y0,   y1,   y2,  1        ]
//   acc    = -2*(x.y) + ||x_m||^2
//   d      = acc + ||y_n||^2
//
// 32-bit A 16x4 layout: lanes 0-15 hold K=0 (vgpr0), K=1 (vgpr1);
//                       lanes 16-31 hold K=2 (vgpr0), K=3 (vgpr1).
// B 4x16 mirrored: lane = column, lane-half selects the K pair.
// C/D 16x16 f32: lane = column (lane&15), row = vgprIdx + 8*(lane>>4).
__global__ __launch_bounds__(256) void chamfer_dir_kernel(
    const float* __restrict__ X,      // (B, 3, N)
    const float* __restrict__ Y,      // (B, 3, N)
    const float* __restrict__ mask,   // (B, N)
    float* __restrict__ out,
    float scale) {
  const int lane = threadIdx.x & 31;
  const int wave = threadIdx.x >> 5;
  const int tile = blockIdx.x * 8 + wave;     // B * N/16 row tiles total
  const int b    = tile >> 8;                 // 256 row tiles per batch
  const int i0   = (tile & 255) << 4;
  const int half = lane >> 4;                 // 0: K={0,1}, 1: K={2,3}
  const int l15  = lane & 15;

  const float* Xb = X    + (size_t)b * 3 * NPT;
  const float* Yb = Y    + (size_t)b * 3 * NPT;
  const float* Mb = mask + (size_t)b * NPT;

  // ---- A tile: rows i0..i0+15, [-2*x, ||x||^2] (once per wave) ----
  const int m  = i0 + l15;
  const float mx = Mb[m];
  const float x0 = Xb[0 * NPT + m] * mx;
  const float x1 = Xb[1 * NPT + m] * mx;
  const float x2 = Xb[2 * NPT + m] * mx;
  const float xn2 = x0 * x0 + x1 * x1 + x2 * x2;
  v2f a;
  a.x = half ? (-2.0f * x2) : (-2.0f * x0);
  a.y = half ? xn2          : (-2.0f * x1);

  float dmin[8];
#pragma unroll
  for (int v = 0; v < 8; ++v) dmin[v] = 3.4e38f;

  for (int j0 = 0; j0 < NPT; j0 += 16) {
    const int n = j0 + l15;
    const float my = Mb[n];
    const float y0 = Yb[0 * NPT + n] * my;
    const float y1 = Yb[1 * NPT + n] * my;
    const float y2 = Yb[2 * NPT + n] * my;
    const float yn2 = y0 * y0 + y1 * y1 + y2 * y2;
    v2f bm;
    bm.x = half ? y2   : y0;
    bm.y = half ? 1.0f : y1;

    v8f acc = {};
    acc = __builtin_amdgcn_wmma_f32_16x16x4_f32(
        /*neg_a=*/false, a, /*neg_b=*/false, bm,
        /*c_mod=*/(short)0, acc, /*reuse_a=*/false, /*reuse_b=*/false);

#pragma unroll
    for (int v = 0; v < 8; ++v)
      dmin[v] = fminf(dmin[v], acc[v] + yn2);
  }

  // min over columns: xor-reduce across the 16 lanes of each half
#pragma unroll
  for (int xm = 1; xm < 16; xm <<= 1) {
#pragma unroll
    for (int v = 0; v < 8; ++v)
      dmin[v] = fminf(dmin[v], __shfl_xor(dmin[v], xm, 32));
  }

  // lane 0 holds mins of rows 0..7, lane 16 rows 8..15; fold halves, sum rows
  float s = 0.0f;
#pragma unroll
  for (int v = 0; v < 8; ++v) s += dmin[v];
  s += __shfl_xor(s, 16, 32);

  // per-block reduction: 1 atomic per block instead of per wave
  __shared__ float red[8];
  if (lane == 0) red[wave] = s * scale;
  __syncthreads();
  if (threadIdx.x == 0) {
    float t = 0.0f;
#pragma unroll
    for (int w = 0; w < 8; ++w) t += red[w];
    atomicAdd(out, t);
  }
}

extern "C" void kernel_launch(void* const* d_in, const int* in_sizes, int n_in,
                              void* d_out, int out_size, void* d_ws, size_t ws_size,
                              hipStream_t stream) {
  (void)in_sizes; (void)n_in; (void)out_size; (void)d_ws; (void)ws_size;
  const float* inp  = (const float*)d_in[0];
  const float* tgt  = (const float*)d_in[1];
  const float* mask = (const float*)d_in[2];
  float* out = (float*)d_out;

  chamfer_zero_kernel<<<1, 1, 0, stream>>>(out);

  const float scale = 1.0f / ((float)BATCH * (float)NPT);
  const int grid = (BATCH * NPT / 16) / 8;  // 512 blocks, 8 waves each
  // mean(dist1): min over tgt for each inp point
  chamfer_dir_kernel<<<grid, 256, 0, stream>>>(inp, tgt, mask, out, scale);
  // mean(dist2): min over inp for each tgt point (roles swapped)
  chamfer_dir_kernel<<<grid, 256, 0, stream>>>(tgt, inp, mask, out, scale);
}